// GNN_SpMoE_node_69415261438525
// MI455X (gfx1250) — compile-verified
//
#include <hip/hip_runtime.h>
#include <math.h>

#define N_NODES   20000
#define N_EDGES   320000
#define DIM       128
#define NLAYERS   4
#define NEXP      4
#define BOND_ROWS 13
#define MTILES    5                      // 80 nodes per wave; 20000 = 250*80 exactly
#define BN_SCALE  0.9999950000374997f    // 1/sqrt(1+1e-5)

typedef float v2f __attribute__((ext_vector_type(2)));
typedef float v8f __attribute__((ext_vector_type(8)));

// ---------------- utility ----------------
__global__ void k_zero(float* __restrict__ p, int n) {
    int i = blockIdx.x * blockDim.x + threadIdx.x;
    if (i < n) p[i] = 0.0f;
}

// ---------------- atom encoder: h[n,d] = sum_i atom_table[x_idx[n,i], d] ----------------
__global__ void k_atom(const float* __restrict__ tab, const int* __restrict__ xidx,
                       float* __restrict__ h) {
    int t = blockIdx.x * blockDim.x + threadIdx.x;
    if (t >= N_NODES * 32) return;
    int n = t >> 5, d = (t & 31) * 4;
    float4 acc = make_float4(0.f, 0.f, 0.f, 0.f);
#pragma unroll
    for (int i = 0; i < 9; ++i) {
        int r = xidx[n * 9 + i];
        float4 v = *(const float4*)(tab + (size_t)r * DIM + d);
        acc.x += v.x; acc.y += v.y; acc.z += v.z; acc.w += v.w;
    }
    *(float4*)(h + (size_t)n * DIM + d) = acc;
}

// ---------------- degree ----------------
__global__ void k_deg_count(const int* __restrict__ ei, float* __restrict__ deg) {
    int e = blockIdx.x * blockDim.x + threadIdx.x;
    if (e < N_EDGES) atomicAdd(&deg[ei[e]], 1.0f);
}
__global__ void k_deg_fin(float* __restrict__ deg, float* __restrict__ dis,
                          float* __restrict__ invdeg) {
    int i = blockIdx.x * blockDim.x + threadIdx.x;
    if (i >= N_NODES) return;
    float dg = deg[i] + 1.0f;
    dis[i]    = rsqrtf(dg);
    invdeg[i] = 1.0f / dg;
}
__global__ void k_norm(const int* __restrict__ ei, const float* __restrict__ dis,
                       float* __restrict__ nrm) {
    int e = blockIdx.x * blockDim.x + threadIdx.x;
    if (e < N_EDGES) nrm[e] = dis[ei[e]] * dis[ei[N_EDGES + e]];
}

// ---------------- top-2 softmax gating: logits = h @ gw[128,4] ----------------
__global__ void k_gate(const float* __restrict__ h, const float* __restrict__ gw,
                       float* __restrict__ gates) {
    int n = blockIdx.x * blockDim.x + threadIdx.x;
    if (n >= N_NODES) return;
    const float* hr = h + (size_t)n * DIM;
    float a0 = 0.f, a1 = 0.f, a2 = 0.f, a3 = 0.f;
    for (int k = 0; k < DIM; ++k) {
        float hv = hr[k];
        float4 w = *(const float4*)(gw + k * 4);
        a0 += hv * w.x; a1 += hv * w.y; a2 += hv * w.z; a3 += hv * w.w;
    }
    float v[4] = {a0, a1, a2, a3};
    int i1 = 0;
#pragma unroll
    for (int e = 1; e < 4; ++e) if (v[e] > v[i1]) i1 = e;
    int i2 = -1;
#pragma unroll
    for (int e = 0; e < 4; ++e) {
        if (e == i1) continue;
        if (i2 < 0 || v[e] > v[i2]) i2 = e;
    }
    float e2 = __expf(v[i2] - v[i1]);
    float s  = 1.0f + e2;
    float g[4] = {0.f, 0.f, 0.f, 0.f};
    g[i1] = 1.0f / s;
    g[i2] = e2 / s;
    *(float4*)(gates + (size_t)n * 4) = make_float4(g[0], g[1], g[2], g[3]);
}

// ---------------- WMMA f32 GEMM: hw[e] = h @ W[e] + b[e] ----------------
// One wave computes an 80x16 output slab (5 x 16x16 tiles sharing each B operand),
// via 160 x v_wmma_f32_16x16x4_f32. B traffic amortized 5x.
__global__ __launch_bounds__(256) void k_gemm(const float* __restrict__ h,
                                              const float* __restrict__ W,     // [NE,128,128]
                                              const float* __restrict__ bias,  // [NE,128]
                                              float* __restrict__ hw) {        // [NE,N,128]
    int lane = threadIdx.x;            // 0..31
    int m    = lane & 15;
    int hi   = lane >> 4;
    int nodeBase = blockIdx.x * (16 * MTILES);
    int colBase  = (int)threadIdx.y * 16;   // 8 waves cover 8 column tiles
    int e        = blockIdx.z;

    const float* We = W + (size_t)e * DIM * DIM;
    const float* hr = h + (size_t)(nodeBase + m) * DIM;   // subtile t at +t*2048 floats

    v8f c0 = {0.f,0.f,0.f,0.f,0.f,0.f,0.f,0.f};
    v8f c1 = c0, c2 = c0, c3 = c0, c4 = c0;

#pragma unroll 2
    for (int kk = 0; kk < DIM; kk += 4) {
        int k0 = kk + 2 * hi;
        v2f b; b.x = We[(size_t)k0 * DIM + colBase + m];
               b.y = We[(size_t)(k0 + 1) * DIM + colBase + m];
        v2f a0; a0.x = hr[k0];           a0.y = hr[k0 + 1];
        v2f a1; a1.x = hr[2048 + k0];    a1.y = hr[2048 + k0 + 1];
        v2f a2; a2.x = hr[4096 + k0];    a2.y = hr[4096 + k0 + 1];
        v2f a3; a3.x = hr[6144 + k0];    a3.y = hr[6144 + k0 + 1];
        v2f a4; a4.x = hr[8192 + k0];    a4.y = hr[8192 + k0 + 1];
        c0 = __builtin_amdgcn_wmma_f32_16x16x4_f32(false, a0, false, b, (short)0, c0, false, false);
        c1 = __builtin_amdgcn_wmma_f32_16x16x4_f32(false, a1, false, b, (short)0, c1, false, false);
        c2 = __builtin_amdgcn_wmma_f32_16x16x4_f32(false, a2, false, b, (short)0, c2, false, false);
        c3 = __builtin_amdgcn_wmma_f32_16x16x4_f32(false, a3, false, b, (short)0, c3, false, false);
        c4 = __builtin_amdgcn_wmma_f32_16x16x4_f32(false, a4, false, b, (short)0, c4, false, false);
    }

    float bval = bias[e * DIM + colBase + m];
    float* out = hw + (size_t)e * N_NODES * DIM
                    + (size_t)nodeBase * DIM + colBase + m;   // subtile t at +t*2048 floats
    v8f acc[MTILES] = {c0, c1, c2, c3, c4};
#pragma unroll
    for (int t = 0; t < MTILES; ++t) {
#pragma unroll
        for (int j = 0; j < 8; ++j) {
            int rowm = j + 8 * hi;                 // C layout: VGPR j -> row j + 8*hi, col m
            out[(size_t)t * 2048 + (size_t)rowm * DIM] = acc[t][j] + bval;
        }
    }
}

// ---------------- edge messages: agg[e][col] += norm * relu(hw[e][row] + ee) ----------------
// Skips experts whose gate at the TARGET node is exactly zero (their agg is
// multiplied by 0 in the combine step, so this is bit-exact) -> ~2x less traffic.
__global__ __launch_bounds__(256) void k_edge(const float* __restrict__ hw,    // [NE,N,128]
                                              const float* __restrict__ bt,    // layer's [NE,13,128]
                                              const int* __restrict__ ei,      // [2,E]
                                              const int* __restrict__ eattr,   // [E,3]
                                              const float* __restrict__ nrm,   // [E]
                                              const float* __restrict__ gates, // [N,NE]
                                              float* __restrict__ agg) {       // [NE,N,128]
    __shared__ float sbt[NEXP * BOND_ROWS * DIM];   // 26.6 KB of the 320 KB WGP LDS
    for (int i = threadIdx.x; i < NEXP * BOND_ROWS * DIM; i += blockDim.x) sbt[i] = bt[i];
    __syncthreads();

    int wpb   = blockDim.x >> 5;
    int wave  = blockIdx.x * wpb + (threadIdx.x >> 5);
    int nWave = gridDim.x * wpb;
    int lane  = threadIdx.x & 31;
    int d     = lane * 4;

    for (int ed = wave; ed < N_EDGES; ed += nWave) {
        int r = ei[ed];
        int c = ei[N_EDGES + ed];
        float4 g4 = *(const float4*)(gates + (size_t)c * 4);   // wave-uniform
        float gv[4] = {g4.x, g4.y, g4.z, g4.w};
        float nv = nrm[ed];
        int b0 = eattr[ed * 3 + 0] * DIM + d;
        int b1 = eattr[ed * 3 + 1] * DIM + d;
        int b2 = eattr[ed * 3 + 2] * DIM + d;
#pragma unroll
        for (int e = 0; e < NEXP; ++e) {
            if (gv[e] == 0.0f) continue;           // uniform branch: expert unused at target
            const float* sb = sbt + e * BOND_ROWS * DIM;
            float4 t0 = *(const float4*)(sb + b0);
            float4 t1 = *(const float4*)(sb + b1);
            float4 t2 = *(const float4*)(sb + b2);
            float4 hv = *(const float4*)(hw + ((size_t)e * N_NODES + r) * DIM + d);
            float4 msg;
            msg.x = nv * fmaxf(hv.x + t0.x + t1.x + t2.x, 0.f);
            msg.y = nv * fmaxf(hv.y + t0.y + t1.y + t2.y, 0.f);
            msg.z = nv * fmaxf(hv.z + t0.z + t1.z + t2.z, 0.f);
            msg.w = nv * fmaxf(hv.w + t0.w + t1.w + t2.w, 0.f);
            float* ap = agg + ((size_t)e * N_NODES + c) * DIM + d;
            atomicAdd(ap + 0, msg.x);
            atomicAdd(ap + 1, msg.y);
            atomicAdd(ap + 2, msg.z);
            atomicAdd(ap + 3, msg.w);
        }
    }
}

// ---------------- combine: out = sum_e gates[e] * BN(agg + relu(hw+root)*invdeg) ----------------
__global__ void k_combine(const float* __restrict__ hw, const float* __restrict__ agg,
                          const float* __restrict__ gates, const float* __restrict__ invdeg,
                          const float* __restrict__ root,   // layer's [NE,128]
                          const float* __restrict__ gamma,  // layer's [NE,128]
                          const float* __restrict__ beta,   // layer's [NE,128]
                          float* __restrict__ hout, int do_relu) {
    int t = blockIdx.x * blockDim.x + threadIdx.x;
    if (t >= N_NODES * 32) return;
    int n = t >> 5, d = (t & 31) * 4;               // n is wave-uniform
    float4 g4 = *(const float4*)(gates + (size_t)n * 4);
    float gv[4] = {g4.x, g4.y, g4.z, g4.w};
    float id = invdeg[n];
    float4 acc = make_float4(0.f, 0.f, 0.f, 0.f);
#pragma unroll
    for (int e = 0; e < NEXP; ++e) {
        if (gv[e] == 0.0f) continue;                // uniform branch: zero contribution
        float4 hv = *(const float4*)(hw  + ((size_t)e * N_NODES + n) * DIM + d);
        float4 ag = *(const float4*)(agg + ((size_t)e * N_NODES + n) * DIM + d);
        float4 rt = *(const float4*)(root  + e * DIM + d);
        float4 gm = *(const float4*)(gamma + e * DIM + d);
        float4 bt = *(const float4*)(beta  + e * DIM + d);
        float zx = ag.x + fmaxf(hv.x + rt.x, 0.f) * id;
        float zy = ag.y + fmaxf(hv.y + rt.y, 0.f) * id;
        float zz = ag.z + fmaxf(hv.z + rt.z, 0.f) * id;
        float zw = ag.w + fmaxf(hv.w + rt.w, 0.f) * id;
        zx = gm.x * (zx * BN_SCALE) + bt.x;
        zy = gm.y * (zy * BN_SCALE) + bt.y;
        zz = gm.z * (zz * BN_SCALE) + bt.z;
        zw = gm.w * (zw * BN_SCALE) + bt.w;
        acc.x += gv[e] * zx; acc.y += gv[e] * zy;
        acc.z += gv[e] * zz; acc.w += gv[e] * zw;
    }
    if (do_relu) {
        acc.x = fmaxf(acc.x, 0.f); acc.y = fmaxf(acc.y, 0.f);
        acc.z = fmaxf(acc.z, 0.f); acc.w = fmaxf(acc.w, 0.f);
    }
    *(float4*)(hout + (size_t)n * DIM + d) = acc;
}

// ---------------- host launcher ----------------
extern "C" void kernel_launch(void* const* d_in, const int* in_sizes, int n_in,
                              void* d_out, int out_size, void* d_ws, size_t ws_size,
                              hipStream_t stream) {
    const float* atom_table = (const float*)d_in[0];
    const float* gate_w     = (const float*)d_in[1];   // [L,128,4]
    const float* lin_w      = (const float*)d_in[2];   // [L,NE,128,128]
    const float* lin_b      = (const float*)d_in[3];   // [L,NE,128]
    const float* root_emb   = (const float*)d_in[4];   // [L,NE,128]
    const float* bond_table = (const float*)d_in[5];   // [L,NE,13,128]
    const float* bn_gamma   = (const float*)d_in[6];   // [L,NE,128]
    const float* bn_beta    = (const float*)d_in[7];   // [L,NE,128]
    const int*   x_idx      = (const int*)d_in[8];     // [N,9]
    const int*   edge_index = (const int*)d_in[9];     // [2,E]
    const int*   edge_attr  = (const int*)d_in[10];    // [E,3]

    size_t ND = (size_t)N_NODES * DIM;
    float* ws     = (float*)d_ws;
    float* hA     = ws;  ws += ND;
    float* hB     = ws;  ws += ND;
    float* hw     = ws;  ws += (size_t)NEXP * ND;
    float* agg    = ws;  ws += (size_t)NEXP * ND;
    float* gates  = ws;  ws += (size_t)N_NODES * NEXP;
    float* deg    = ws;  ws += N_NODES;
    float* dis    = ws;  ws += N_NODES;
    float* invdeg = ws;  ws += N_NODES;
    float* nrm    = ws;  ws += N_EDGES;

    dim3 b256(256);

    k_atom<<<(N_NODES * 32 + 255) / 256, b256, 0, stream>>>(atom_table, x_idx, hA);
    k_zero<<<(N_NODES + 255) / 256, b256, 0, stream>>>(deg, N_NODES);
    k_deg_count<<<(N_EDGES + 255) / 256, b256, 0, stream>>>(edge_index, deg);
    k_deg_fin<<<(N_NODES + 255) / 256, b256, 0, stream>>>(deg, dis, invdeg);
    k_norm<<<(N_EDGES + 255) / 256, b256, 0, stream>>>(edge_index, dis, nrm);

    float* hcur = hA;
    float* hnext = hB;
    for (int layer = 0; layer < NLAYERS; ++layer) {
        k_gate<<<(N_NODES + 255) / 256, b256, 0, stream>>>(
            hcur, gate_w + (size_t)layer * DIM * NEXP, gates);

        dim3 gg(N_NODES / (16 * MTILES), 1, NEXP), gb(32, 8);
        k_gemm<<<gg, gb, 0, stream>>>(
            hcur, lin_w + (size_t)layer * NEXP * DIM * DIM,
            lin_b + (size_t)layer * NEXP * DIM, hw);

        int aggN = NEXP * (int)ND;
        k_zero<<<(aggN + 255) / 256, b256, 0, stream>>>(agg, aggN);

        k_edge<<<2048, b256, 0, stream>>>(
            hw, bond_table + (size_t)layer * NEXP * BOND_ROWS * DIM,
            edge_index, edge_attr, nrm, gates, agg);

        float* dst = (layer == NLAYERS - 1) ? (float*)d_out : hnext;
        k_combine<<<(N_NODES * 32 + 255) / 256, b256, 0, stream>>>(
            hw, agg, gates, invdeg,
            root_emb + (size_t)layer * NEXP * DIM,
            bn_gamma + (size_t)layer * NEXP * DIM,
            bn_beta  + (size_t)layer * NEXP * DIM,
            dst, (layer < NLAYERS - 1) ? 1 : 0);

        float* tmp = hcur; hcur = hnext; hnext = tmp;
    }
}